// PositionalTopDownHTMM_9732395893087
// MI455X (gfx1250) — compile-verified
//
#include <hip/hip_runtime.h>
#include <math.h>

typedef __attribute__((ext_vector_type(2))) float v2f;
typedef __attribute__((ext_vector_type(8))) float v8f;

#define TK 4          // children per node
#define TC 8          // hidden states
#define TG 8          // groups
#define TM 256        // emission alphabet
#define TN 87381      // total nodes
#define TSL 21845     // start of leaf level
#define TCG 64        // C*G per node

__device__ __forceinline__ int aidx(int i, int j, int k, int g) {
    // A[i,j,k,g], shape (8,8,4,8) row-major
    return ((i * TC + j) * TK + k) * TG + g;
}

// ---------------- setup: softmaxes + A*logA ----------------
__global__ void htmm_setup(const float* __restrict__ A, const float* __restrict__ B,
                           const float* __restrict__ Pi,
                           float* __restrict__ smA, float* __restrict__ alogA,
                           float* __restrict__ smB, float* __restrict__ smPi,
                           float* __restrict__ logPi) {
    int tid = threadIdx.x; // 256 threads
    { // A: softmax over i per column (j,k,g); 256 columns; elem (i,col) at A[col + i*256]
        int col = tid;
        float x[TC], e[TC];
        float mx = -1e30f;
        for (int i = 0; i < TC; ++i) { x[i] = A[col + i * 256]; mx = fmaxf(mx, x[i]); }
        float s = 0.f;
        for (int i = 0; i < TC; ++i) { e[i] = expf(x[i] - mx); s += e[i]; }
        float ls = logf(s);
        for (int i = 0; i < TC; ++i) {
            float sm = e[i] / s;
            smA[col + i * 256] = sm;
            alogA[col + i * 256] = sm * ((x[i] - mx) - ls); // smA * log(smA)
        }
    }
    if (tid < TC * TG) { // B: softmax over m per (c,g)
        int c = tid >> 3, g = tid & 7;
        const float* src = B + c * (TM * TG) + g;
        float* dst = smB + c * (TM * TG) + g;
        float mx = -1e30f;
        for (int m = 0; m < TM; ++m) mx = fmaxf(mx, src[m * TG]);
        float s = 0.f;
        for (int m = 0; m < TM; ++m) s += expf(src[m * TG] - mx);
        for (int m = 0; m < TM; ++m) dst[m * TG] = expf(src[m * TG] - mx) / s;
    }
    if (tid < TG) { // Pi: softmax over c per g
        int g = tid;
        float x[TC], e[TC];
        float mx = -1e30f;
        for (int c = 0; c < TC; ++c) { x[c] = Pi[c * TG + g]; mx = fmaxf(mx, x[c]); }
        float s = 0.f;
        for (int c = 0; c < TC; ++c) { e[c] = expf(x[c] - mx); s += e[c]; }
        float ls = logf(s);
        for (int c = 0; c < TC; ++c) {
            smPi[c * TG + g] = e[c] / s;
            logPi[c * TG + g] = (x[c] - mx) - ls;
        }
    }
}

__global__ void htmm_copy64(float* __restrict__ dst, const float* __restrict__ src) {
    if (threadIdx.x < TCG) dst[threadIdx.x] = src[threadIdx.x];
}

// ------------- downward prior: pc[i,n] = sum_j A[i,j]*pp[j,n], per (k,g) -------------
__global__ void htmm_down(const float* __restrict__ smA, float* __restrict__ prior,
                          int sp, int npar, int sc) {
    int kg = blockIdx.y; int k = kg >> 3, g = kg & 7;
    int lane = threadIdx.x;
    int half = lane >> 4, m = lane & 15;
    int node = blockIdx.x * 16 + m;
    bool nv = node < npar;
    // A frag: M=i=m (rows >=8 zero), Kdim=j: lanes 0-15 hold K={0,1}, lanes 16-31 K={2,3}
    float a00 = 0.f, a01 = 0.f, a10 = 0.f, a11 = 0.f;
    if (m < TC) {
        int j0 = half * 2;
        a00 = smA[aidx(m, j0,     k, g)];
        a01 = smA[aidx(m, j0 + 1, k, g)];
        a10 = smA[aidx(m, j0 + 4, k, g)];
        a11 = smA[aidx(m, j0 + 5, k, g)];
    }
    // B frag: Kdim=j rows, N=node cols
    float b00 = 0.f, b01 = 0.f, b10 = 0.f, b11 = 0.f;
    if (nv) {
        const float* pp = prior + (size_t)(sp + node) * TCG + g;
        int j0 = half * 2;
        b00 = pp[(j0)     * 8]; b01 = pp[(j0 + 1) * 8];
        b10 = pp[(j0 + 4) * 8]; b11 = pp[(j0 + 5) * 8];
    }
    v8f acc = {0.f,0.f,0.f,0.f,0.f,0.f,0.f,0.f};
    v2f A0 = {a00, a01}, B0 = {b00, b01};
    acc = __builtin_amdgcn_wmma_f32_16x16x4_f32(false, A0, false, B0, (short)0, acc, false, false);
    v2f A1 = {a10, a11}, B1 = {b10, b11};
    acc = __builtin_amdgcn_wmma_f32_16x16x4_f32(false, A1, false, B1, (short)0, acc, false, false);
    if (half == 0 && nv) { // rows i=0..7 live in lanes 0-15
        float* pc = prior + (size_t)(sc + node * TK + k) * TCG + g;
        #pragma unroll
        for (int i = 0; i < TC; ++i) pc[i * 8] = acc[i];
    }
}

// ------------- beta = prior * B[:,label], leaves normalized over C -------------
__global__ void htmm_beta_init(const float* __restrict__ prior, const float* __restrict__ smB,
                               const int* __restrict__ labels, float* __restrict__ beta) {
    int wid = (int)((blockIdx.x * blockDim.x + threadIdx.x) >> 5);
    int lane = threadIdx.x & 31;
    if (wid >= TN) return;
    int lab = labels[wid];
    int c0 = lane >> 3, g = lane & 7, c1 = c0 + 4;
    float p0 = prior[(size_t)wid * TCG + c0 * 8 + g] * smB[c0 * (TM * TG) + lab * TG + g];
    float p1 = prior[(size_t)wid * TCG + c1 * 8 + g] * smB[c1 * (TM * TG) + lab * TG + g];
    if (wid >= TSL) {
        float s = p0 + p1;
        s += __shfl_xor(s, 8, 32);
        s += __shfl_xor(s, 16, 32);
        p0 /= s; p1 /= s;
    }
    beta[(size_t)wid * TCG + c0 * 8 + g] = p0;
    beta[(size_t)wid * TCG + c1 * 8 + g] = p1;
}

// ------------- upward: buv[j,n]=sum_i A[i,j]*(beta/prior)[i,n]; combine over k -------------
__global__ void __launch_bounds__(1024) htmm_up(
    const float* __restrict__ smA, float* __restrict__ beta,
    const float* __restrict__ prior, float* __restrict__ tbeta,
    int sp, int npar, int sc) {
    __shared__ float sbuv[32 * 8 * 16]; // [kg][j][n]
    __shared__ float sred[1024];
    int tid = threadIdx.x;
    int kg = tid >> 5; int k = kg >> 3, g = kg & 7;
    int lane = tid & 31; int half = lane >> 4, m = lane & 15;
    int node = blockIdx.x * 16 + m;
    bool nv = node < npar;
    // A frag (transposed use): M=j=m, Kdim=i
    float a00 = 0.f, a01 = 0.f, a10 = 0.f, a11 = 0.f;
    if (m < TC) {
        int i0 = half * 2;
        a00 = smA[aidx(i0,     m, k, g)];
        a01 = smA[aidx(i0 + 1, m, k, g)];
        a10 = smA[aidx(i0 + 4, m, k, g)];
        a11 = smA[aidx(i0 + 5, m, k, g)];
    }
    // B frag: Kdim=i rows of bp = beta/prior at child
    float b00 = 0.f, b01 = 0.f, b10 = 0.f, b11 = 0.f;
    if (nv) {
        size_t ch = (size_t)(sc + node * TK + k) * TCG + g;
        const float* bb = beta + ch;
        const float* pp = prior + ch;
        int i0 = half * 2;
        b00 = bb[i0 * 8]       / pp[i0 * 8];
        b01 = bb[(i0 + 1) * 8] / pp[(i0 + 1) * 8];
        b10 = bb[(i0 + 4) * 8] / pp[(i0 + 4) * 8];
        b11 = bb[(i0 + 5) * 8] / pp[(i0 + 5) * 8];
    }
    v8f acc = {0.f,0.f,0.f,0.f,0.f,0.f,0.f,0.f};
    v2f A0 = {a00, a01}, B0 = {b00, b01};
    acc = __builtin_amdgcn_wmma_f32_16x16x4_f32(false, A0, false, B0, (short)0, acc, false, false);
    v2f A1 = {a10, a11}, B1 = {b10, b11};
    acc = __builtin_amdgcn_wmma_f32_16x16x4_f32(false, A1, false, B1, (short)0, acc, false, false);
    if (half == 0) {
        #pragma unroll
        for (int j = 0; j < TC; ++j) sbuv[(kg * 8 + j) * 16 + m] = acc[j];
        if (nv) {
            float* tb = tbeta + (size_t)(sc + node * TK + k) * TCG + g;
            #pragma unroll
            for (int j = 0; j < TC; ++j) tb[j * 8] = acc[j];
        }
    }
    __syncthreads();
    // combine: bu[jg] = beta_parent[jg] * prod_k buv[k]; normalize over all 64 (C,G)
    int nloc = tid >> 6, jg = tid & 63;
    int j2 = jg >> 3, g2 = jg & 7;
    int node2 = blockIdx.x * 16 + nloc;
    float bu = 0.f;
    if (node2 < npar) {
        float p = 1.f;
        #pragma unroll
        for (int kk = 0; kk < TK; ++kk) p *= sbuv[((kk * 8 + g2) * 8 + j2) * 16 + nloc];
        bu = beta[(size_t)(sp + node2) * TCG + jg] * p;
    }
    sred[tid] = bu;
    __syncthreads();
    #pragma unroll
    for (int off = 32; off >= 1; off >>= 1) {
        if ((tid & 63) < off) sred[tid] += sred[tid + off];
        __syncthreads();
    }
    if (node2 < npar) beta[(size_t)(sp + node2) * TCG + jg] = bu / sred[tid & ~63];
}

// ------------- eps downward (E-step) + A-loglik partials -------------
__global__ void htmm_eps(const float* __restrict__ smA, const float* __restrict__ alogA,
                         const float* __restrict__ beta, const float* __restrict__ prior,
                         const float* __restrict__ tbeta, float* __restrict__ epsb,
                         float* __restrict__ Apart,
                         int sp, int npar, int sc, int apBase) {
    int kg = blockIdx.y; int k = kg >> 3, g = kg & 7;
    int lane = threadIdx.x; int half = lane >> 4, m = lane & 15;
    int node = blockIdx.x * 16 + m;
    bool nv = node < npar;
    // Two A frags: smA and smA*log(smA); M=i=m, Kdim=j
    float a00=0.f,a01=0.f,a10=0.f,a11=0.f, l00=0.f,l01=0.f,l10=0.f,l11=0.f;
    if (m < TC) {
        int j0 = half * 2;
        a00 = smA[aidx(m, j0,     k, g)]; l00 = alogA[aidx(m, j0,     k, g)];
        a01 = smA[aidx(m, j0 + 1, k, g)]; l01 = alogA[aidx(m, j0 + 1, k, g)];
        a10 = smA[aidx(m, j0 + 4, k, g)]; l10 = alogA[aidx(m, j0 + 4, k, g)];
        a11 = smA[aidx(m, j0 + 5, k, g)]; l11 = alogA[aidx(m, j0 + 5, k, g)];
    }
    // B frag: v[j,n] = eps_parent[j]/tbeta_child[j]
    float b00 = 0.f, b01 = 0.f, b10 = 0.f, b11 = 0.f;
    if (nv) {
        const float* ep = epsb  + (size_t)(sp + node) * TCG + g;
        const float* tb = tbeta + (size_t)(sc + node * TK + k) * TCG + g;
        int j0 = half * 2;
        b00 = ep[j0 * 8]       / tb[j0 * 8];
        b01 = ep[(j0 + 1) * 8] / tb[(j0 + 1) * 8];
        b10 = ep[(j0 + 4) * 8] / tb[(j0 + 4) * 8];
        b11 = ep[(j0 + 5) * 8] / tb[(j0 + 5) * 8];
    }
    v2f A0 = {a00,a01}, A1 = {a10,a11}, L0 = {l00,l01}, L1 = {l10,l11};
    v2f B0 = {b00,b01}, B1 = {b10,b11};
    v8f w = {0.f,0.f,0.f,0.f,0.f,0.f,0.f,0.f};
    w = __builtin_amdgcn_wmma_f32_16x16x4_f32(false, A0, false, B0, (short)0, w, false, false);
    w = __builtin_amdgcn_wmma_f32_16x16x4_f32(false, A1, false, B1, (short)0, w, false, false);
    v8f u = {0.f,0.f,0.f,0.f,0.f,0.f,0.f,0.f};
    u = __builtin_amdgcn_wmma_f32_16x16x4_f32(false, L0, false, B0, (short)0, u, false, false);
    u = __builtin_amdgcn_wmma_f32_16x16x4_f32(false, L1, false, B1, (short)0, u, false, false);
    float alh = 0.f;
    if (half == 0 && nv) { // each lane 0-15 holds full i-column for its node
        size_t ch = (size_t)(sc + node * TK + k) * TCG + g;
        float num[TC]; float s = 0.f;
        #pragma unroll
        for (int i = 0; i < TC; ++i) {
            float cf = beta[ch + i * 8] / prior[ch + i * 8];
            num[i] = cf * w[i];
            s += num[i];
            alh += cf * u[i];
        }
        float inv = 1.f / s;
        #pragma unroll
        for (int i = 0; i < TC; ++i) epsb[ch + i * 8] = num[i] * inv;
    }
    #pragma unroll
    for (int off = 16; off >= 1; off >>= 1) alh += __shfl_xor(alh, off, 32);
    if (lane == 0) Apart[apBase + blockIdx.y * gridDim.x + blockIdx.x] = alh;
}

// ------------- B log-likelihood partials (deterministic) -------------
__global__ void htmm_blh(const float* __restrict__ epsb, const float* __restrict__ smB,
                         const int* __restrict__ labels, float* __restrict__ Bpart) {
    __shared__ float sm[256];
    int tid = threadIdx.x;
    float acc = 0.f;
    const int total = TN * TCG;
    for (int e = blockIdx.x * blockDim.x + tid; e < total; e += gridDim.x * blockDim.x) {
        int n = e >> 6; int cg = e & 63; int c = cg >> 3, g = cg & 7;
        int lab = labels[n];
        acc += epsb[e] * logf(smB[c * (TM * TG) + lab * TG + g]);
    }
    sm[tid] = acc;
    __syncthreads();
    for (int off = 128; off >= 1; off >>= 1) {
        if (tid < off) sm[tid] += sm[tid + off];
        __syncthreads();
    }
    if (tid == 0) Bpart[blockIdx.x] = sm[0];
}

// ------------- final deterministic reduction: out = -(Pi_lh + A_lh + B_lh) -------------
__global__ void htmm_final(const float* __restrict__ Apart, int nA,
                           const float* __restrict__ Bpart, int nB,
                           const float* __restrict__ epsb, const float* __restrict__ logPi,
                           float* __restrict__ out) {
    __shared__ float sm[256];
    int tid = threadIdx.x;
    float acc = 0.f;
    for (int i = tid; i < nA; i += 256) acc += Apart[i];
    for (int i = tid; i < nB; i += 256) acc += Bpart[i];
    if (tid < TCG) acc += epsb[tid] * logPi[tid]; // Pi_lh from eps at root
    sm[tid] = acc;
    __syncthreads();
    for (int off = 128; off >= 1; off >>= 1) {
        if (tid < off) sm[tid] += sm[tid + off];
        __syncthreads();
    }
    if (tid == 0) out[0] = -sm[0];
}

extern "C" void kernel_launch(void* const* d_in, const int* in_sizes, int n_in,
                              void* d_out, int out_size, void* d_ws, size_t ws_size,
                              hipStream_t stream) {
    const float* A      = (const float*)d_in[0];
    const float* B      = (const float*)d_in[1];
    const float* Pi     = (const float*)d_in[2];
    const int*   labels = (const int*)d_in[3];
    float* out = (float*)d_out;
    float* ws  = (float*)d_ws;

    static const int START_[9] = {0, 1, 5, 21, 85, 341, 1365, 5461, 21845};
    static const int SIZE_[8]  = {1, 4, 16, 64, 256, 1024, 4096, 16384};
    static const int TILES_[8] = {1, 1, 1, 4, 16, 64, 256, 1024};
    static const int APBASE_[8]= {0, 32, 64, 96, 224, 736, 2784, 10976};
    const int TOT_AP = 32 * 1367; // 43744
    const int NBLH = 512;

    const size_t NCG = (size_t)TN * TCG;
    const size_t OFF_SMA   = 0;
    const size_t OFF_ALOGA = OFF_SMA + 2048;
    const size_t OFF_SMB   = OFF_ALOGA + 2048;
    const size_t OFF_SMPI  = OFF_SMB + (size_t)TC * TM * TG;
    const size_t OFF_LOGPI = OFF_SMPI + TCG;
    const size_t OFF_PRIOR = OFF_LOGPI + TCG;
    const size_t OFF_BETA  = OFF_PRIOR + NCG;
    const size_t OFF_TBETA = OFF_BETA + NCG;
    const size_t OFF_EPS   = OFF_TBETA + NCG;
    const size_t OFF_APART = OFF_EPS + NCG;
    const size_t OFF_BPART = OFF_APART + TOT_AP;
    const size_t WS_NEED   = OFF_BPART + NBLH;
    if (ws_size < WS_NEED * sizeof(float)) return;

    float* smA   = ws + OFF_SMA;
    float* alogA = ws + OFF_ALOGA;
    float* smB   = ws + OFF_SMB;
    float* smPi  = ws + OFF_SMPI;
    float* logPi = ws + OFF_LOGPI;
    float* prior = ws + OFF_PRIOR;
    float* beta  = ws + OFF_BETA;
    float* tbeta = ws + OFF_TBETA;
    float* epsb  = ws + OFF_EPS;
    float* Apart = ws + OFF_APART;
    float* Bpart = ws + OFF_BPART;

    htmm_setup<<<1, 256, 0, stream>>>(A, B, Pi, smA, alogA, smB, smPi, logPi);

    for (int t = 0; t < 2; ++t) {
        const int* lab = labels + (size_t)t * TN;

        htmm_copy64<<<1, 64, 0, stream>>>(prior, smPi);
        for (int l = 0; l < 8; ++l) {
            htmm_down<<<dim3(TILES_[l], 32), 32, 0, stream>>>(
                smA, prior, START_[l], SIZE_[l], START_[l + 1]);
        }

        htmm_beta_init<<<(TN + 7) / 8, 256, 0, stream>>>(prior, smB, lab, beta);

        for (int l = 7; l >= 0; --l) {
            htmm_up<<<TILES_[l], 1024, 0, stream>>>(
                smA, beta, prior, tbeta, START_[l], SIZE_[l], START_[l + 1]);
        }

        htmm_copy64<<<1, 64, 0, stream>>>(epsb, beta); // eps[root] = beta[root]

        for (int l = 0; l < 8; ++l) {
            htmm_eps<<<dim3(TILES_[l], 32), 32, 0, stream>>>(
                smA, alogA, beta, prior, tbeta, epsb, Apart,
                START_[l], SIZE_[l], START_[l + 1], APBASE_[l]);
        }

        htmm_blh<<<NBLH, 256, 0, stream>>>(epsb, smB, lab, Bpart);
        htmm_final<<<1, 256, 0, stream>>>(Apart, TOT_AP, Bpart, NBLH, epsb, logPi, out + t);
    }
    (void)in_sizes; (void)n_in; (void)out_size;
}